// EdgeNetwork_37280316129537
// MI455X (gfx1250) — compile-verified
//
#include <hip/hip_runtime.h>

typedef __attribute__((ext_vector_type(2))) float v2f;
typedef __attribute__((ext_vector_type(8))) float v8f;

#define N_PAIRS 64000
#define N_ATOMS 16000
#define NPF     8
#define NH      100
#define NT      7      // 7 n-tiles of 16 -> 112 cols (>=100, padded)
#define NBLK    9      // 8 W feature blocks + 1 bias block
#define KSTEPS  25     // 100 / 4 (K=4 per f32 WMMA)
#define TILE_M  16
#define WPB     8      // waves per block

// ---------------------------------------------------------------------------
// Pack Bex (900 x 112) into WMMA-fragment order in d_ws:
//   fragment F = f*25+ks, n-tile t, lane l, vgpr v  ->  Bfrag[((F*7+t)*32+l)*2+v]
//   value = Bex[f*100 + 4*ks + 2*(l>=16) + v,  t*16 + (l&15)]
//   Bex[(f,j), i] = W[f, i*100 + j]   (f<8) ;  Bex[(8,j), i] = b[i*100 + j]
// ---------------------------------------------------------------------------
__global__ void prep_bfrag(const float* __restrict__ W, const float* __restrict__ b,
                           float* __restrict__ Bfrag) {
  int gid = blockIdx.x * blockDim.x + threadIdx.x;
  const int total = NBLK * KSTEPS * NT * 64;
  if (gid >= total) return;
  int F    = gid / (NT * 64);
  int rem  = gid % (NT * 64);
  int t    = rem / 64;
  int e    = rem % 64;
  int lane = e >> 1;
  int v    = e & 1;
  int f    = F / KSTEPS;
  int ks   = F % KSTEPS;
  int hi   = lane >> 4;
  int krel = 4 * ks + 2 * hi + v;      // j within block, 0..99
  int col  = t * 16 + (lane & 15);     // i, 0..111
  float val = 0.0f;
  if (col < NH) {
    val = (f < NPF) ? W[f * (NH * NH) + col * NH + krel]
                    : b[col * NH + krel];
  }
  Bfrag[gid] = val;
}

__global__ void zero_out(float* __restrict__ out, int n) {
  int gid = blockIdx.x * blockDim.x + threadIdx.x;
  if (gid < n) out[gid] = 0.0f;
}

// ---------------------------------------------------------------------------
// One wave32 per 16-edge tile: fused gather + GEMM (f32 WMMA) + atomic scatter
// ---------------------------------------------------------------------------
__global__ __launch_bounds__(256) void edge_gemm(
    const float* __restrict__ pair_features,
    const float* __restrict__ atom_features,
    const int*   __restrict__ atom_to_pair,
    const float* __restrict__ Bfrag,
    float*       __restrict__ out) {
  __shared__ float lds[WPB * (TILE_M * NH + TILE_M * NBLK)];

  const int lane = threadIdx.x & 31;
  const int wid  = threadIdx.x >> 5;
  const int tile = blockIdx.x * WPB + wid;
  const int base = tile * TILE_M;

  float* lsrc = lds + wid * (TILE_M * NH + TILE_M * NBLK);
  float* lpf  = lsrc + TILE_M * NH;

  // stage pair features; block 8 gets implicit weight 1.0 (bias block)
  for (int idx = lane; idx < TILE_M * NBLK; idx += 32) {
    int r = idx / NBLK;
    int f = idx % NBLK;
    lpf[idx] = (f < NPF) ? pair_features[(base + r) * NPF + f] : 1.0f;
  }
  // stage gathered source-atom rows (16 x 100, stride 100 -> conflict-free)
  for (int r = 0; r < TILE_M; ++r) {
    int asrc = atom_to_pair[(base + r) * 2 + 1];
    const float* srow = atom_features + (long)asrc * NH;
    for (int c = lane; c < NH; c += 32)
      lsrc[r * NH + c] = srow[c];
  }
  __syncthreads();

  const int mrow = lane & 15;
  const int hi   = lane >> 4;
  const int koff = 2 * hi;          // f32 A-frag: lanes 16-31 hold K+2,K+3

  v8f acc[NT] = {};

  for (int f = 0; f < NBLK; ++f) {
    float w = lpf[mrow * NBLK + f];
    for (int ks = 0; ks < KSTEPS; ++ks) {
      int k0 = 4 * ks + koff;
      v2f a;
      a.x = w * lsrc[mrow * NH + k0];
      a.y = w * lsrc[mrow * NH + k0 + 1];
      const float* bp = Bfrag + (long)((f * KSTEPS + ks) * NT) * 64 + lane * 2;
#pragma unroll
      for (int t = 0; t < NT; ++t) {
        v2f bb = *(const v2f*)(bp + t * 64);
        acc[t] = __builtin_amdgcn_wmma_f32_16x16x4_f32(
            false, a, false, bb, (short)0, acc[t], false, false);
      }
    }
  }

  // epilogue: segment-sum via atomic scatter-add
#pragma unroll
  for (int t = 0; t < NT; ++t) {
    int col = t * 16 + mrow;
    if (col < NH) {
#pragma unroll
      for (int r = 0; r < 8; ++r) {
        int M   = r + 8 * hi;
        int dst = atom_to_pair[(base + M) * 2];
        atomicAdd(out + (long)dst * NH + col, acc[t][r]);
      }
    }
  }
}

extern "C" void kernel_launch(void* const* d_in, const int* in_sizes, int n_in,
                              void* d_out, int out_size, void* d_ws, size_t ws_size,
                              hipStream_t stream) {
  const float* pf  = (const float*)d_in[0];   // (64000, 8)   f32
  const float* af  = (const float*)d_in[1];   // (16000, 100) f32
  const int*   a2p = (const int*)  d_in[2];   // (64000, 2)   i32
  const float* W   = (const float*)d_in[3];   // (8, 10000)   f32
  const float* b   = (const float*)d_in[4];   // (10000,)     f32
  float* out   = (float*)d_out;               // (16000, 100) f32
  float* Bfrag = (float*)d_ws;                // 100800 floats (~394 KB)

  const int bfragN = NBLK * KSTEPS * NT * 64;
  prep_bfrag<<<(bfragN + 255) / 256, 256, 0, stream>>>(W, b, Bfrag);
  zero_out<<<(out_size + 255) / 256, 256, 0, stream>>>(out, out_size);

  const int tiles = N_PAIRS / TILE_M;                 // 4000
  edge_gemm<<<tiles / WPB, WPB * 32, 0, stream>>>(pf, af, a2p, Bfrag, out);
}